// GraphAttention_40613210751279
// MI455X (gfx1250) — compile-verified
//
#include <hip/hip_runtime.h>
#include <cmath>

typedef __attribute__((ext_vector_type(2))) float v2f;
typedef __attribute__((ext_vector_type(8))) float v8f;

#define NNODES 8192
#define DIM    512
#define NEG_BIG 1000000000.0f

// ---------------------------------------------------------------------------
// Kernel 1: h = x @ W^T   (M=8192, N=512, K=512), fp32 WMMA 16x16x4.
// Also writes hT (512 x 8192) so that the big G@h GEMM can load its B
// fragments as contiguous float2 (k-major).
// A[16x4] frag: lanes 0-15 hold M=lane, K = k+0 (v0), k+1 (v1);
//               lanes16-31 hold M=lane-16, K = k+2 (v0), k+3 (v1).
// B[4x16] frag: per-lane N = lane&15; K pairs contiguous since B = W^T.
// ---------------------------------------------------------------------------
__global__ __launch_bounds__(256) void gat_xw_kernel(
    const float* __restrict__ x, const float* __restrict__ W,
    float* __restrict__ h, float* __restrict__ hT) {
  const int lane = threadIdx.x & 31;
  const int wave = threadIdx.x >> 5;
  const int l15  = lane & 15;
  const int half = lane >> 4;          // 0 or 1
  const int m_base = blockIdx.x * 16;
  const int n_base = (blockIdx.y * 8 + wave) * 16;

  const float* xrow = x + (m_base + l15) * DIM;   // A: row = M
  const float* wrow = W + (n_base + l15) * DIM;   // B: row = N (B[k][n]=W[n][k])

  v8f acc = {};
  for (int k = 0; k < DIM; k += 4) {
    const int kk = k + half * 2;
    v2f a = *(const v2f*)(xrow + kk);
    v2f b = *(const v2f*)(wrow + kk);
    acc = __builtin_amdgcn_wmma_f32_16x16x4_f32(
        false, a, false, b, (short)0, acc, false, false);
  }
  // D layout: VGPR r -> M = m_base + r + half*8, N = n_base + l15
#pragma unroll
  for (int r = 0; r < 8; ++r) {
    h[(m_base + r + half * 8) * DIM + n_base + l15] = acc[r];
  }
  // Transposed copy: per lane, col fixed -> 8 contiguous floats in hT.
  float* htp = hT + (size_t)(n_base + l15) * NNODES + m_base + half * 8;
#pragma unroll
  for (int r = 0; r < 8; ++r) {
    htp[r] = acc[r];
  }
}

// ---------------------------------------------------------------------------
// Kernel 2: s_i[n] = dot(h[n], a1), s_j[n] = dot(h[n], a2). One wave per row.
// ---------------------------------------------------------------------------
__global__ __launch_bounds__(256) void gat_scores_kernel(
    const float* __restrict__ h, const float* __restrict__ a,
    float* __restrict__ s_i, float* __restrict__ s_j) {
  const int lane = threadIdx.x & 31;
  const int wave = threadIdx.x >> 5;
  const int row  = blockIdx.x * 8 + wave;
  const float* hrow = h + row * DIM;

  float acc1 = 0.f, acc2 = 0.f;
  for (int k = lane; k < DIM; k += 32) {
    const float hv = hrow[k];
    acc1 += hv * a[k];
    acc2 += hv * a[DIM + k];
  }
#pragma unroll
  for (int off = 16; off > 0; off >>= 1) {
    acc1 += __shfl_down(acc1, off, 32);
    acc2 += __shfl_down(acc2, off, 32);
  }
  if (lane == 0) {
    s_i[row] = acc1;
    s_j[row] = acc2;
  }
}

// ---------------------------------------------------------------------------
// Kernel 3: row-wise masked softmax. One block (256 thr) per row; the 8192-
// element e-row lives in LDS (32 KB of the 320 KB WGP LDS), so adj is read
// once and G is written once (normalized).
// ---------------------------------------------------------------------------
__global__ __launch_bounds__(256) void gat_softmax_kernel(
    const int* __restrict__ adj, const float* __restrict__ s_i,
    const float* __restrict__ s_j, float* __restrict__ G) {
  __shared__ float ebuf[NNODES];
  __shared__ float red[256];

  const int i   = blockIdx.x;
  const int tid = threadIdx.x;
  const float si = s_i[i];
  const int*  arow = adj + (size_t)i * NNODES;

  // Phase 1: compute masked e into LDS, track local max
  float lmax = -3.4e38f;
  for (int j = tid; j < NNODES; j += 256) {
    const float v  = si + s_j[j];
    const float lr = (v > 0.f) ? v : 0.2f * v;               // leaky_relu(0.2)
    const float m  = ((arow[j] > 0) ? 1.f : 0.f) + ((j == i) ? 1.f : 0.f);
    const float pen = (m < 1.f) ? (1.f - m) : 0.f;           // clip(1-m, 0, inf)
    const float e  = lr * m - NEG_BIG * pen;
    ebuf[j] = e;
    lmax = fmaxf(lmax, e);
  }
  red[tid] = lmax;
  __syncthreads();
  for (int s = 128; s > 0; s >>= 1) {
    if (tid < s) red[tid] = fmaxf(red[tid], red[tid + s]);
    __syncthreads();
  }
  const float rowmax = red[0];
  __syncthreads();

  // Phase 2: exponentiate in LDS, accumulate sum
  float lsum = 0.f;
  for (int j = tid; j < NNODES; j += 256) {
    const float p = __expf(ebuf[j] - rowmax);
    ebuf[j] = p;
    lsum += p;
  }
  red[tid] = lsum;
  __syncthreads();
  for (int s = 128; s > 0; s >>= 1) {
    if (tid < s) red[tid] += red[tid + s];
    __syncthreads();
  }
  const float inv = 1.f / red[0];

  // Phase 3: single normalized write of G row
  float* grow = G + (size_t)i * NNODES;
  for (int j = tid; j < NNODES; j += 256) {
    grow[j] = ebuf[j] * inv;
  }
}

// ---------------------------------------------------------------------------
// Kernel 4: out = elu(G @ h)  (M=8192, K=8192, N=512), fp32 WMMA 16x16x4.
// B fragments come from hT (k-major), so both A and B are single b64 loads:
// 5 global_load_b64 per 4 WMMAs. Each wave: 16x64 output (4 accumulators),
// A-frag (G) reused 4x per k-step. 8 waves/block cover all 512 columns, and
// hT (16 MB) stays L2-resident so G is the only HBM stream.
// ---------------------------------------------------------------------------
__global__ __launch_bounds__(256) void gat_gh_kernel(
    const float* __restrict__ G, const float* __restrict__ hT,
    float* __restrict__ out) {
  const int lane = threadIdx.x & 31;
  const int wave = threadIdx.x >> 5;
  const int l15  = lane & 15;
  const int half = lane >> 4;
  const int m_base = blockIdx.x * 16;
  const int n0 = wave * 64;

  const float* grow = G + (size_t)(m_base + l15) * NNODES;       // A: row = M
  const float* htp  = hT + (size_t)(n0 + l15) * NNODES;          // B: row = N

  v8f acc0 = {}, acc1 = {}, acc2 = {}, acc3 = {};
  for (int k = 0; k < NNODES; k += 4) {
    const int kk = k + half * 2;
    v2f a  = *(const v2f*)(grow + kk);
    v2f b0 = *(const v2f*)(htp + kk);                            // n = n0+l15
    v2f b1 = *(const v2f*)(htp + (size_t)16 * NNODES + kk);      // +16 cols
    v2f b2 = *(const v2f*)(htp + (size_t)32 * NNODES + kk);      // +32 cols
    v2f b3 = *(const v2f*)(htp + (size_t)48 * NNODES + kk);      // +48 cols
    acc0 = __builtin_amdgcn_wmma_f32_16x16x4_f32(false, a, false, b0, (short)0, acc0, false, false);
    acc1 = __builtin_amdgcn_wmma_f32_16x16x4_f32(false, a, false, b1, (short)0, acc1, false, false);
    acc2 = __builtin_amdgcn_wmma_f32_16x16x4_f32(false, a, false, b2, (short)0, acc2, false, false);
    acc3 = __builtin_amdgcn_wmma_f32_16x16x4_f32(false, a, false, b3, (short)0, acc3, false, false);
  }

#pragma unroll
  for (int r = 0; r < 8; ++r) {
    const int row = m_base + r + half * 8;
    float* orow = out + (size_t)row * DIM;
    float v;
    v = acc0[r]; orow[n0 +      l15] = (v > 0.f) ? v : (__expf(v) - 1.f);
    v = acc1[r]; orow[n0 + 16 + l15] = (v > 0.f) ? v : (__expf(v) - 1.f);
    v = acc2[r]; orow[n0 + 32 + l15] = (v > 0.f) ? v : (__expf(v) - 1.f);
    v = acc3[r]; orow[n0 + 48 + l15] = (v > 0.f) ? v : (__expf(v) - 1.f);
  }
}

// ---------------------------------------------------------------------------
extern "C" void kernel_launch(void* const* d_in, const int* in_sizes, int n_in,
                              void* d_out, int out_size, void* d_ws, size_t ws_size,
                              hipStream_t stream) {
  const float* x   = (const float*)d_in[0];   // [8192, 512]
  const int*   adj = (const int*)  d_in[1];   // [8192, 8192]
  const float* W   = (const float*)d_in[2];   // [512, 512]
  const float* a   = (const float*)d_in[3];   // [1, 1024]

  float* out = (float*)d_out;                 // [8192, 512]
  float* G   = out + (size_t)NNODES * DIM;    // [8192, 8192] (second output)

  float* h   = (float*)d_ws;                  // [8192, 512] = 16 MB
  float* hT  = h + (size_t)NNODES * DIM;      // [512, 8192] = 16 MB
  float* s_i = hT + (size_t)NNODES * DIM;     // [8192]
  float* s_j = s_i + NNODES;                  // [8192]

  gat_xw_kernel<<<dim3(NNODES / 16, DIM / 128), 256, 0, stream>>>(x, W, h, hT);
  gat_scores_kernel<<<NNODES / 8, 256, 0, stream>>>(h, a, s_i, s_j);
  gat_softmax_kernel<<<NNODES, 256, 0, stream>>>(adj, s_i, s_j, G);
  gat_gh_kernel<<<NNODES / 16, 256, 0, stream>>>(G, hT, out);
}